// YoloV1Loss_5136780886770
// MI455X (gfx1250) — compile-verified
//
#include <hip/hip_runtime.h>
#include <math.h>
#include <stdint.h>

// Problem constants (from reference)
#define SG 7
#define NB 2          // boxes per cell
#define NC 20         // classes
#define NG 8          // gt boxes
#define BATCH 16384
#define CELL 30       // NB*5 + NC
#define CELLS 49      // SG*SG
#define FPB 1470      // CELLS*CELL floats per batch image
#define PAIRS 735     // FPB/2, 8-byte units per image
#define PAIRS_PAD 768 // 12 * 64 lanes
#define LAMBDA_COORD 5.0f
#define LAMBDA_NOOBJ 0.5f
#define EPSI 1e-6f

__device__ __forceinline__ float iou1(float px, float py, float pw, float ph,
                                      float gx, float gy, float gw, float gh) {
    float px1 = px - pw * 0.5f, px2 = px + pw * 0.5f;
    float py1 = py - ph * 0.5f, py2 = py + ph * 0.5f;
    float gx1 = gx - gw * 0.5f, gx2 = gx + gw * 0.5f;
    float gy1 = gy - gh * 0.5f, gy2 = gy + gh * 0.5f;
    float iw = fminf(px2, gx2) - fmaxf(px1, gx1);
    float ih = fminf(py2, gy2) - fmaxf(py1, gy1);
    iw = iw > 0.0f ? iw : 0.0f;
    ih = ih > 0.0f ? ih : 0.0f;
    float inter = iw * ih;
    float uni = pw * ph + gw * gh - inter;
    return inter / (uni + EPSI);
}

__global__ __launch_bounds__(64) void yolo_cell_kernel(
    const float* __restrict__ outputs,
    const float* __restrict__ gt_boxes,
    const int* __restrict__ gt_labels,
    float* __restrict__ partial) {

    __shared__ __align__(16) float s_out[PAIRS_PAD * 2]; // 1536 floats (1470 used)
    __shared__ float s_gt[NG * 4];
    __shared__ int s_lab[NG];
    __shared__ float s_red[64];

    const int b = blockIdx.x;
    const int tid = threadIdx.x;

    const float* gbase = outputs + (size_t)b * FPB;

    // CDNA5 prefetch hint (global_prefetch_b8): warm the streamed image data.
    __builtin_prefetch(gbase + tid * 8, 0, 1);

    // --- Async global -> LDS staging of this image's 1470 floats (b64/lane).
    // Batch byte stride = 5880 (8B aligned); indices clamped so EXEC is all-1s.
    {
        const unsigned lds_base = (unsigned)(uintptr_t)(void*)s_out; // LDS byte offset
        const uint64_t sbase = (uint64_t)(uintptr_t)gbase;
#pragma unroll
        for (int i = 0; i < 12; ++i) {
            int pair = tid + i * 64;             // 0..767
            int gp = pair > (PAIRS - 1) ? (PAIRS - 1) : pair; // clamp global side
            unsigned voff = (unsigned)gp * 8u;   // byte offset from SADDR
            unsigned ldsa = lds_base + (unsigned)pair * 8u;
            asm volatile("global_load_async_to_lds_b64 %0, %1, %2"
                         :
                         : "v"(ldsa), "v"(voff), "s"(sbase)
                         : "memory");
        }
    }

    // Small per-image ground-truth data via regular loads.
    if (tid < NG * 4) s_gt[tid] = gt_boxes[(size_t)b * (NG * 4) + tid];
    if (tid < NG) s_lab[tid] = gt_labels[(size_t)b * NG + tid];

    // Wait for this wave's async transfers, then make LDS visible block-wide.
    asm volatile("s_wait_asynccnt 0x0" ::: "memory");
    __syncthreads();

    float loss = 0.0f;
    if (tid < CELLS) {
        const float* cell = s_out + tid * CELL;

        float px[NB], py[NB], pw[NB], ph[NB], pc[NB];
#pragma unroll
        for (int k = 0; k < NB; ++k) {
            px[k] = cell[k * 5 + 0];
            py[k] = cell[k * 5 + 1];
            pw[k] = cell[k * 5 + 2];
            ph[k] = cell[k * 5 + 3];
            pc[k] = cell[k * 5 + 4];
        }

        // Fused max/argmax over n (first-occurrence ties, like jnp.argmax),
        // capturing iou[b=0..1] at j_star.
        float best = -1.0f;
        float at0 = 0.0f, at1 = 0.0f;
#pragma unroll
        for (int n = 0; n < NG; ++n) {
            float gx = s_gt[n * 4 + 0], gy = s_gt[n * 4 + 1];
            float gw = s_gt[n * 4 + 2], gh = s_gt[n * 4 + 3];
            float i0 = iou1(px[0], py[0], pw[0], ph[0], gx, gy, gw, gh);
            float i1 = iou1(px[1], py[1], pw[1], ph[1], gx, gy, gw, gh);
            float m = fmaxf(i0, i1);
            if (m > best) { best = m; at0 = i0; at1 = i1; }
        }
        const int bsel = (at1 > at0) ? 1 : 0; // argmax over B, first on ties

        // NOTE: reference indexes gt_boxes' N axis with best_b (0 or 1).
        float gx = s_gt[bsel * 4 + 0], gy = s_gt[bsel * 4 + 1];
        float gw = s_gt[bsel * 4 + 2], gh = s_gt[bsel * 4 + 3];

        float dx = px[bsel] - gx;
        float dy = py[bsel] - gy;
        float dw = sqrtf(pw[bsel]) - sqrtf(gw);
        float dh = sqrtf(ph[bsel]) - sqrtf(gh);
        float loc = LAMBDA_COORD * (dx * dx + dy * dy + dw * dw + dh * dh);

        float dconf = pc[bsel] - best;
        float conf_obj = dconf * dconf;

        // log-softmax over 20 classes (in LDS at cell+10)
        const float* cls = cell + NB * 5;
        float mx = cls[0];
#pragma unroll
        for (int c = 1; c < NC; ++c) mx = fmaxf(mx, cls[c]);
        float se = 0.0f;
#pragma unroll
        for (int c = 0; c < NC; ++c) se += expf(cls[c] - mx);
        float lse = mx + logf(se);

        float ce = 0.0f;
#pragma unroll
        for (int n = 0; n < NG; ++n) ce += lse - cls[s_lab[n]];
        ce *= (1.0f / (float)NG);

        float noobj = LAMBDA_NOOBJ * (pc[0] * pc[0] + pc[1] * pc[1]);

        loss = (best > 0.0f) ? (loc + conf_obj + ce) : noobj;
    }

    // Deterministic fixed-order block tree reduction (64 -> 1).
    s_red[tid] = loss;
    __syncthreads();
#pragma unroll
    for (int off = 32; off > 0; off >>= 1) {
        if (tid < off) s_red[tid] += s_red[tid + off];
        __syncthreads();
    }
    if (tid == 0) partial[b] = s_red[0];
}

__global__ __launch_bounds__(256) void yolo_reduce_kernel(
    const float* __restrict__ partial, float* __restrict__ out) {
    __shared__ double sred[256];
    const int tid = threadIdx.x;
    double acc = 0.0;
    for (int i = tid; i < BATCH; i += 256) acc += (double)partial[i];
    sred[tid] = acc;
    __syncthreads();
#pragma unroll
    for (int off = 128; off > 0; off >>= 1) {
        if (tid < off) sred[tid] += sred[tid + off];
        __syncthreads();
    }
    if (tid == 0) out[0] = (float)(sred[0] / (double)BATCH);
}

extern "C" void kernel_launch(void* const* d_in, const int* in_sizes, int n_in,
                              void* d_out, int out_size, void* d_ws, size_t ws_size,
                              hipStream_t stream) {
    const float* outputs = (const float*)d_in[0];   // (BATCH, 7, 7, 30) fp32
    const float* gt_boxes = (const float*)d_in[1];  // (BATCH, 8, 4)  fp32
    const int* gt_labels = (const int*)d_in[2];     // (BATCH, 8)     int32
    float* partial = (float*)d_ws;                  // BATCH floats (64 KB)

    yolo_cell_kernel<<<BATCH, 64, 0, stream>>>(outputs, gt_boxes, gt_labels, partial);
    yolo_reduce_kernel<<<1, 256, 0, stream>>>(partial, (float*)d_out);
}